// GraphAttention_74517682585647
// MI455X (gfx1250) — compile-verified
//
#include <hip/hip_runtime.h>

#define B_SZ   2
#define N_SZ   512
#define K_IN   128
#define D_EMB  64
#define ALPHA_ 0.2f
#define NEG_BIG -9.0e15f

typedef float v2f __attribute__((ext_vector_type(2)));
typedef float v8f __attribute__((ext_vector_type(8)));

// ---------------------------------------------------------------------------
// Kernel 1: hidden[b, n, :] = x[b, n, :] @ W   via V_WMMA_F32_16X16X4_F32
// One wave (32 lanes) per 16-row tile; 4 column tiles of 16; K=128 in steps of 4.
// A-frag: lane L -> x[row0 + (L&15), k + 2*(L>>4) + {0,1}]
// B-frag: lane L -> W[k + 2*(L>>4) + {0,1}, ct*16 + (L&15)]
// ---------------------------------------------------------------------------
__global__ __launch_bounds__(32) void gat_hidden_kernel(
    const float* __restrict__ x, const float* __restrict__ W,
    float* __restrict__ hidden) {
  const int tile = blockIdx.x;            // B * (N/16) = 64 tiles
  const int b    = tile >> 5;             // N/16 == 32
  const int row0 = (tile & 31) << 4;
  const int lane = threadIdx.x & 31;
  const int half = lane >> 4;
  const int l16  = lane & 15;

  const float* xr = x + (((size_t)b * N_SZ) + row0 + l16) * K_IN;

  v8f acc0 = {}, acc1 = {}, acc2 = {}, acc3 = {};

  for (int k = 0; k < K_IN; k += 4) {
    const int kk = k + 2 * half;
    v2f af;
    af.x = xr[kk];
    af.y = xr[kk + 1];

    const float* wp = W + (size_t)kk * D_EMB + l16;
    v2f bf0, bf1, bf2, bf3;
    bf0.x = wp[0];  bf0.y = wp[D_EMB + 0];
    bf1.x = wp[16]; bf1.y = wp[D_EMB + 16];
    bf2.x = wp[32]; bf2.y = wp[D_EMB + 32];
    bf3.x = wp[48]; bf3.y = wp[D_EMB + 48];

    acc0 = __builtin_amdgcn_wmma_f32_16x16x4_f32(false, af, false, bf0, (short)0, acc0, false, false);
    acc1 = __builtin_amdgcn_wmma_f32_16x16x4_f32(false, af, false, bf1, (short)0, acc1, false, false);
    acc2 = __builtin_amdgcn_wmma_f32_16x16x4_f32(false, af, false, bf2, (short)0, acc2, false, false);
    acc3 = __builtin_amdgcn_wmma_f32_16x16x4_f32(false, af, false, bf3, (short)0, acc3, false, false);
  }

  // C/D layout: VGPR r, lane L -> (row = r + 8*half, col = l16 + 16*ct)
  float* hr = hidden + (((size_t)b * N_SZ) + row0) * D_EMB;
#pragma unroll
  for (int r = 0; r < 8; ++r) {
    const int mrow = r + 8 * half;
    hr[(size_t)mrow * D_EMB + l16]      = acc0[r];
    hr[(size_t)mrow * D_EMB + 16 + l16] = acc1[r];
    hr[(size_t)mrow * D_EMB + 32 + l16] = acc2[r];
    hr[(size_t)mrow * D_EMB + 48 + l16] = acc3[r];
  }
}

// ---------------------------------------------------------------------------
// Kernel 2: s1[b,n] = hidden[b,n,:] . a[0:64],  s2[b,n] = hidden[b,n,:] . a[64:128]
// ---------------------------------------------------------------------------
__global__ void gat_scores_kernel(const float* __restrict__ hidden,
                                  const float* __restrict__ avec,
                                  float* __restrict__ s1, float* __restrict__ s2) {
  const int idx = blockIdx.x * blockDim.x + threadIdx.x;
  if (idx >= B_SZ * N_SZ) return;
  const float* h = hidden + (size_t)idx * D_EMB;
  float t1 = 0.f, t2 = 0.f;
#pragma unroll 8
  for (int d = 0; d < D_EMB; ++d) {
    const float hv = h[d];
    t1 = fmaf(hv, avec[d], t1);
    t2 = fmaf(hv, avec[D_EMB + d], t2);
  }
  s1[idx] = t1;
  s2[idx] = t2;
}

// ---------------------------------------------------------------------------
// Kernel 3: fused masked-softmax + (attention @ hidden) + bias.
// Logit structure (from the reference's cat(dim=1)+view index algebra):
//   i <  N/2 : logit = s1[2i + (j>=N/2)] + s2[2i + (j>=N/2)]
//   i >= N/2 : logit = s1[(2j)%N] + s2[(2j)%N + 1]
// leaky-relu, mask adj==0 -> -9e15, softmax over j, then P @ hidden via WMMA.
// One wave per 16-row tile. Pass A: per-row online (max, sum). Pass B: WMMA.
// ---------------------------------------------------------------------------
__global__ __launch_bounds__(32) void gat_attn_out_kernel(
    const float* __restrict__ hidden, const int* __restrict__ adj,
    const float* __restrict__ s1, const float* __restrict__ s2,
    const float* __restrict__ bias, float* __restrict__ out) {
  const int tile = blockIdx.x;            // B * (N/16)
  const int b    = tile >> 5;
  const int row0 = (tile & 31) << 4;
  const int lane = threadIdx.x & 31;
  const int half = lane >> 4;
  const int l16  = lane & 15;
  const int irow = row0 + l16;            // this lane's attention row

  const float* s1b  = s1 + b * N_SZ;
  const float* s2b  = s2 + b * N_SZ;
  const int*   adjr = adj + (((size_t)b * N_SZ) + irow) * N_SZ;
  const float* hb   = hidden + (size_t)b * N_SZ * D_EMB;

  const bool lowI = (row0 < (N_SZ / 2));  // uniform across the tile
  float llA = 0.f, llB = 0.f;
  if (lowI) {
    const float ta = s1b[2 * irow]     + s2b[2 * irow];
    const float tb = s1b[2 * irow + 1] + s2b[2 * irow + 1];
    llA = (ta >= 0.f) ? ta : ALPHA_ * ta;   // leaky-relu pre-applied
    llB = (tb >= 0.f) ? tb : ALPHA_ * tb;
  }

  auto masked_logit = [&](int j) -> float {
    float lg;
    if (lowI) {
      lg = (j < (N_SZ / 2)) ? llA : llB;
    } else {
      const int rr = (2 * j) & (N_SZ - 1);
      const float t = s1b[rr] + s2b[rr + 1];
      lg = (t >= 0.f) ? t : ALPHA_ * t;
    }
    return (adjr[j] > 0) ? lg : NEG_BIG;
  };

  // ---- Pass A: online max & sum-of-exp for row irow (half=0 evens, half=1 odds)
  float m = NEG_BIG, ssum = 0.f;
  for (int j = half; j < N_SZ; j += 2) {
    const float v = masked_logit(j);
    if (v > m) { ssum *= __expf(m - v); m = v; }
    ssum += __expf(v - m);
  }
  {
    const float mo = __shfl_xor(m, 16);
    const float so = __shfl_xor(ssum, 16);
    const float mn = fmaxf(m, mo);
    ssum = ssum * __expf(m - mn) + so * __expf(mo - mn);
    m = mn;                                 // row max, same in both halves
  }
  const float linv = 1.0f / ssum;           // softmax denominator^-1 for row irow

  // ---- Pass B: out_tile += P(16xN) @ hidden(NxD) via WMMA, K=4 per step
  v8f acc0 = {}, acc1 = {}, acc2 = {}, acc3 = {};
  for (int j0 = 0; j0 < N_SZ; j0 += 4) {
    const int jA = j0 + 2 * half;
    v2f pa;                                 // A-frag: P[irow, jA], P[irow, jA+1]
    pa.x = __expf(masked_logit(jA)     - m);
    pa.y = __expf(masked_logit(jA + 1) - m);

    const float* hp = hb + (size_t)jA * D_EMB + l16;  // B-frag rows jA, jA+1
    v2f bf0, bf1, bf2, bf3;
    bf0.x = hp[0];  bf0.y = hp[D_EMB + 0];
    bf1.x = hp[16]; bf1.y = hp[D_EMB + 16];
    bf2.x = hp[32]; bf2.y = hp[D_EMB + 32];
    bf3.x = hp[48]; bf3.y = hp[D_EMB + 48];

    acc0 = __builtin_amdgcn_wmma_f32_16x16x4_f32(false, pa, false, bf0, (short)0, acc0, false, false);
    acc1 = __builtin_amdgcn_wmma_f32_16x16x4_f32(false, pa, false, bf1, (short)0, acc1, false, false);
    acc2 = __builtin_amdgcn_wmma_f32_16x16x4_f32(false, pa, false, bf2, (short)0, acc2, false, false);
    acc3 = __builtin_amdgcn_wmma_f32_16x16x4_f32(false, pa, false, bf3, (short)0, acc3, false, false);
  }

  // ---- normalize rows and add bias; lane r<16 holds linv for row r
  float* outr = out + (((size_t)b * N_SZ) + row0) * D_EMB;
#pragma unroll
  for (int r = 0; r < 8; ++r) {
    const int mrow = r + 8 * half;
    const float sc = __shfl(linv, mrow);    // row mrow's 1/l from lane mrow
    outr[(size_t)mrow * D_EMB + l16]      = acc0[r] * sc + bias[l16];
    outr[(size_t)mrow * D_EMB + 16 + l16] = acc1[r] * sc + bias[16 + l16];
    outr[(size_t)mrow * D_EMB + 32 + l16] = acc2[r] * sc + bias[32 + l16];
    outr[(size_t)mrow * D_EMB + 48 + l16] = acc3[r] * sc + bias[48 + l16];
  }
}

// ---------------------------------------------------------------------------
extern "C" void kernel_launch(void* const* d_in, const int* in_sizes, int n_in,
                              void* d_out, int out_size, void* d_ws, size_t ws_size,
                              hipStream_t stream) {
  const float* x    = (const float*)d_in[0];   // [2,512,128]
  const int*   adj  = (const int*)d_in[1];     // [2,512,512]
  const float* W    = (const float*)d_in[2];   // [128,64]
  const float* avec = (const float*)d_in[3];   // [128,1]
  const float* bias = (const float*)d_in[4];   // [64]
  float* out = (float*)d_out;                  // [2,512,64]

  float* hidden = (float*)d_ws;                          // B*N*D floats (256 KB)
  float* s1 = hidden + (size_t)B_SZ * N_SZ * D_EMB;      // B*N floats
  float* s2 = s1 + (size_t)B_SZ * N_SZ;                  // B*N floats

  const dim3 grid(B_SZ * (N_SZ / 16));   // 64 waves
  const dim3 blk(32);                    // one wave32 per workgroup

  gat_hidden_kernel<<<grid, blk, 0, stream>>>(x, W, hidden);
  gat_scores_kernel<<<(B_SZ * N_SZ + 255) / 256, 256, 0, stream>>>(hidden, avec, s1, s2);
  gat_attn_out_kernel<<<grid, blk, 0, stream>>>(hidden, adj, s1, s2, bias, out);
}